// DistillLoss_75548474737114
// MI455X (gfx1250) — compile-verified
//
#include <hip/hip_runtime.h>
#include <math.h>

#ifndef __has_builtin
#define __has_builtin(x) 0
#endif

// ---------------- compile-time config ----------------
constexpr int   THREADS = 256;           // 8 waves of 32 on CDNA5
constexpr int   WAVES   = THREADS / 32;
constexpr int   VMAX    = 10016;         // LDS staging capacity (floats), >= V=10000
constexpr float LSM_C   = 0.1f;
constexpr float SOFTW_C = 0.5f;

typedef int   v4i __attribute__((ext_vector_type(4)));
typedef float v4f __attribute__((ext_vector_type(4)));

// ---------------- wave32 reductions ----------------
__device__ __forceinline__ float waveMax(float v) {
#pragma unroll
  for (int o = 16; o > 0; o >>= 1) v = fmaxf(v, __shfl_xor(v, o, 32));
  return v;
}
__device__ __forceinline__ float waveSum(float v) {
#pragma unroll
  for (int o = 16; o > 0; o >>= 1) v += __shfl_xor(v, o, 32);
  return v;
}

// One block per (b,t) token row: blockIdx.x = t, blockIdx.y = b (no int div).
// Computes soft/hard per-token losses and accumulates masked sums into
// acc[0] (soft) and acc[1] (hard).
__global__ __launch_bounds__(THREADS)
void distill_main_kernel(const float* __restrict__ logits,
                         const int*   __restrict__ ys,
                         const float* __restrict__ soft,
                         const int*   __restrict__ ylens,
                         float*       __restrict__ acc,
                         int T, int V) {
  __shared__ alignas(16) float s_logits[VMAX];
  __shared__ float s_m[WAVES];
  __shared__ float s_p[WAVES][4];

  const int t = blockIdx.x;
  const int b = blockIdx.y;
  if (t >= ylens[b]) return;               // masked token: skip all traffic

  const int row  = b * T + t;
  const int tid  = threadIdx.x;
  const int lane = tid & 31;
  const int wid  = tid >> 5;

  const float* __restrict__ lrow = logits + (size_t)row * (size_t)V;
  const float* __restrict__ srow = soft   + (size_t)row * (size_t)V;
  const int nvec = V >> 2;                 // float4 chunks (V=10000 -> 2500)

  // ---- Stage logits row into LDS. CDNA5 path: async global->LDS b128
  //      (fire-and-forget DMA tracked on ASYNCcnt; no VGPR return traffic). ----
#if __has_builtin(__builtin_amdgcn_global_load_async_to_lds_b128)
  for (int i = tid; i < nvec; i += THREADS) {
    __builtin_amdgcn_global_load_async_to_lds_b128(
        (__attribute__((address_space(1))) v4i*)(lrow + 4 * i),
        (__attribute__((address_space(3))) v4i*)(&s_logits[4 * i]),
        /*offset=*/0, /*cpol=*/0);
  }
  for (int i = (nvec << 2) + tid; i < V; i += THREADS) s_logits[i] = lrow[i];
#if __has_builtin(__builtin_amdgcn_s_wait_asynccnt)
  __builtin_amdgcn_s_wait_asynccnt(0);
#else
  asm volatile("s_wait_asynccnt 0" ::: "memory");
#endif
#else
  // Fallback: vectorized load + LDS store
  for (int i = tid; i < nvec; i += THREADS)
    reinterpret_cast<v4f*>(s_logits)[i] =
        reinterpret_cast<const v4f*>(lrow)[i];
  for (int i = (nvec << 2) + tid; i < V; i += THREADS) s_logits[i] = lrow[i];
#endif
  __syncthreads();

  // ---- Pass A: row max (from LDS) ----
  float m = -INFINITY;
  for (int i = tid; i < nvec; i += THREADS) {
    v4f x = reinterpret_cast<const v4f*>(s_logits)[i];
    m = fmaxf(m, fmaxf(fmaxf(x.x, x.y), fmaxf(x.z, x.w)));
  }
  for (int i = (nvec << 2) + tid; i < V; i += THREADS) m = fmaxf(m, s_logits[i]);
  m = waveMax(m);
  if (lane == 0) s_m[wid] = m;
  __syncthreads();
  if (tid == 0) {
    float M = s_m[0];
#pragma unroll
    for (int w = 1; w < WAVES; ++w) M = fmaxf(M, s_m[w]);
    s_m[0] = M;
  }
  __syncthreads();
  const float M = s_m[0];

  // ---- Pass B: sum exp / sum logits / dot(logits,soft) / sum soft ----
  // logits re-read from LDS (free of HBM); soft streams once, non-temporal
  // (read-once working set exceeds the 192 MB L2 -> don't thrash it).
  float e = 0.0f, sl = 0.0f, dt = 0.0f, ss = 0.0f;
  for (int i = tid; i < nvec; i += THREADS) {
    v4f x = reinterpret_cast<const v4f*>(s_logits)[i];
    v4f p = __builtin_nontemporal_load(reinterpret_cast<const v4f*>(srow) + i);
    e  += __expf(x.x - M) + __expf(x.y - M) + __expf(x.z - M) + __expf(x.w - M);
    sl += (x.x + x.y) + (x.z + x.w);
    dt  = fmaf(x.x, p.x, fmaf(x.y, p.y, fmaf(x.z, p.z, fmaf(x.w, p.w, dt))));
    ss += (p.x + p.y) + (p.z + p.w);
  }
  for (int i = (nvec << 2) + tid; i < V; i += THREADS) {
    float x = s_logits[i], p = srow[i];
    e += __expf(x - M); sl += x; dt = fmaf(x, p, dt); ss += p;
  }
  e  = waveSum(e);
  sl = waveSum(sl);
  dt = waveSum(dt);
  ss = waveSum(ss);
  if (lane == 0) { s_p[wid][0] = e; s_p[wid][1] = sl; s_p[wid][2] = dt; s_p[wid][3] = ss; }
  __syncthreads();

  if (tid == 0) {
    float E = 0.0f, SL = 0.0f, DT = 0.0f, SS = 0.0f;
#pragma unroll
    for (int w = 0; w < WAVES; ++w) {
      E += s_p[w][0]; SL += s_p[w][1]; DT += s_p[w][2]; SS += s_p[w][3];
    }
    const float lse      = M + logf(E);
    const float soft_tok = DT - lse * SS;                 // sum_v lp*soft
    const float lp_sum   = SL - (float)V * lse;           // sum_v lp
    const int   y        = ys[row];
    const float lp_y     = s_logits[y] - lse;
    const float hard_tok = (1.0f - LSM_C) * lp_y +
                           (LSM_C / (float)(V - 1)) * (lp_sum - lp_y);
    atomicAdd(&acc[0], soft_tok);
    atomicAdd(&acc[1], hard_tok);
  }
}

__global__ __launch_bounds__(THREADS)
void distill_zero_kernel(float* __restrict__ acc) {
  if (threadIdx.x < 2) acc[threadIdx.x] = 0.0f;
}

__global__ void distill_final_kernel(const float* __restrict__ acc,
                                     float* __restrict__ out, float invB) {
  if (blockIdx.x == 0 && threadIdx.x == 0) {
    const float loss_soft = -acc[0] * invB;
    const float loss_hard = -acc[1] * invB;
    out[0] = SOFTW_C * loss_soft + (1.0f - SOFTW_C) * loss_hard;
    out[1] = loss_soft;
    out[2] = loss_hard;
  }
}

// ---------------- launcher ----------------
extern "C" void kernel_launch(void* const* d_in, const int* in_sizes, int n_in,
                              void* d_out, int out_size, void* d_ws, size_t ws_size,
                              hipStream_t stream) {
  (void)n_in; (void)out_size; (void)ws_size;

  const float* logits = (const float*)d_in[0];   // [B,T,V] f32
  const int*   ys     = (const int*)  d_in[1];   // [B,T]   i32 (JAX x64 off)
  const float* soft   = (const float*)d_in[2];   // [B,T,V] f32
  const int*   ylens  = (const int*)  d_in[3];   // [B]     i32
  float*       out    = (float*)d_out;           // 3 scalars
  float*       acc    = (float*)d_ws;            // 2 accumulators

  const int n_tok = in_sizes[1];                 // B*T
  const int B     = in_sizes[3];
  const int T     = n_tok / B;
  const int V     = in_sizes[0] / n_tok;         // 10000 (<= VMAX)

  distill_zero_kernel<<<1, THREADS, 0, stream>>>(acc);
  distill_main_kernel<<<dim3(T, B), THREADS, 0, stream>>>(logits, ys, soft,
                                                          ylens, acc, T, V);
  distill_final_kernel<<<1, 64, 0, stream>>>(acc, out, 1.0f / (float)B);
}